// DTCWTLayerQshift_Inv_30253749633567
// MI455X (gfx1250) — compile-verified
//
#include <hip/hip_runtime.h>
#include <hip/hip_bf16.h>

// CDNA5 / gfx1250: wave32, f32 WMMA 16x16x4.
typedef __attribute__((ext_vector_type(2))) float v2f;
typedef __attribute__((ext_vector_type(8))) float v8f;

// ---------------------------------------------------------------------------
// Filter constants (qshift 10-tap).  _H0B = reverse(_H0A);
// _H1A = _H0B * (-1)^k ; _H1B = reverse(_H1A).
// Pair0 (G0B,G0A) = (_H0A, _H0B)   -> sum(ha*hb) > 0
// Pair1 (G1B,G1A) = (_H1B, _H1A)   -> sum(ha*hb) < 0
// ---------------------------------------------------------------------------
__constant__ double c_H0A[10] = {
    0.0351638365171441, 0.0, -0.0883294244510729, 0.233890320607236,
    0.760272369066126,  0.587518297723561, 0.0, -0.114301837144249, 0.0, 0.0};

__device__ __forceinline__ int reflect_idx(int x) {
  // reflect(x, -0.5, 127.5) for integer x, period 256 (x >= -256 here)
  int p = x & 255;
  return (p < 128) ? p : 255 - p;
}

// Build dense 256x128 filter matrices M (out[n] = sum_src M[n][src]*in[src])
// reproducing _colifilt's phase split, ta/tb selection, tap reversal and
// boundary reflection for r = 128.
__global__ __launch_bounds__(256) void build_M(float* __restrict__ M0,
                                               float* __restrict__ M1) {
  const int n = threadIdx.x;  // output row 0..255
  for (int pair = 0; pair < 2; ++pair) {
    float* M = pair ? M1 : M0;
    double ha[10], hb[10];
    for (int k = 0; k < 10; ++k) {
      double h0a = c_H0A[k];
      double h0b = c_H0A[9 - k];
      if (pair == 0) { ha[k] = h0a; hb[k] = h0b; }
      else {
        double s = (k & 1) ? -1.0 : 1.0;
        ha[k] = -h0a * s;  // _H1B[k]
        hb[k] =  h0b * s;  // _H1A[k]
      }
    }
    double dot = 0.0;
    for (int k = 0; k < 10; ++k) dot += ha[k] * hb[k];
    const int ta_off = (dot > 0.0) ? 0 : -1;
    const int tb_off = (dot > 0.0) ? -1 : 0;

    const int i = n >> 2, j = n & 3;
    // j=0: (Xb, hao)  j=1: (Xa, hbo)  j=2: (Xb, hae)  j=3: (Xa, hbe)
    const bool useA = (j == 1) || (j == 3);
    const int off = useA ? ta_off : tb_off;
    double h5[5];
    for (int k = 0; k < 5; ++k) {
      if      (j == 0) h5[k] = ha[2 * k];
      else if (j == 1) h5[k] = hb[2 * k];
      else if (j == 2) h5[k] = ha[2 * k + 1];
      else             h5[k] = hb[2 * k + 1];
    }
    float* row = &M[n * 128];
    for (int k = 0; k < 128; ++k) row[k] = 0.0f;
    for (int k = 0; k < 5; ++k) {
      int s = i + k;                          // t[s] = 2 + 2s
      int src = reflect_idx(2 + 2 * s + off - 5);  // xe[t+off]
      row[src] += (float)h5[4 - k];           // hf = h[::-1]
    }
  }
}

// ---------------------------------------------------------------------------
// c2q gather: element pair (h, col0), (h, col0+1) of a 128x128 c2q image built
// from subbands p,q.  col0 is even.  Branchless (EXEC must stay all-ones).
// ---------------------------------------------------------------------------
__device__ __forceinline__ v2f c2q_pair(const float* __restrict__ Yhr,
                                        const float* __restrict__ Yhi,
                                        int bc, int p, int q, int h, int col0) {
  const int hp = h >> 1, wp = col0 >> 1;
  const size_t base = (size_t)bc * 6 * 4096;  // 64*64 per band
  const size_t o1 = base + (size_t)p * 4096 + (size_t)hp * 64 + wp;
  const size_t o2 = base + (size_t)q * 4096 + (size_t)hp * 64 + wp;
  const float w1r = Yhr[o1], w2r = Yhr[o2];
  const float w1i = Yhi[o1], w2i = Yhi[o2];
  const float s = 0.70710678118654752440f;
  const bool odd = (h & 1) != 0;
  v2f r;
  r.x = (odd ? (w1i - w2i) : (w1r + w2r)) * s;  // even output col
  r.y = (odd ? (w2r - w1r) : (w1i + w2i)) * s;  // odd output col
  return r;
}

#define WMMA_F32(A, B, C) \
  __builtin_amdgcn_wmma_f32_16x16x4_f32(false, (A), false, (B), (short)0, (C), false, false)

// ---------------------------------------------------------------------------
// Pass 1: row filter.  One wave -> one 16x16 tile of BOTH y1 and y2.
//   y1[h,wo] = sum_wi Yl[h,wi]*M0[wo,wi] + HL[h,wi]*M1[wo,wi]
//   y2[h,wo] = sum_wi LH[h,wi]*M0[wo,wi] + HH[h,wi]*M1[wo,wi]
// Band structure: support of outputs [wo0,wo0+15] lies in the 16-wide window
// kw = clamp(wo0/2-4, 0, 112).
// ---------------------------------------------------------------------------
__global__ __launch_bounds__(256)
void pass1_rowfilt(const float* __restrict__ Yl, const float* __restrict__ Yhr,
                   const float* __restrict__ Yhi, const float* __restrict__ M0,
                   const float* __restrict__ M1, float* __restrict__ Y1,
                   float* __restrict__ Y2) {
  const int lane = threadIdx.x & 31;
  const int wv   = threadIdx.x >> 5;
  const int tile = blockIdx.x * 8 + wv;      // 512*8*16 tiles total
  const int wob = tile & 15;
  const int hb  = (tile >> 4) & 7;
  const int bc  = tile >> 7;                 // 0..511
  const int wo0 = wob * 16, h0 = hb * 16;
  int kw = wo0 / 2 - 4;
  kw = kw < 0 ? 0 : (kw > 112 ? 112 : kw);

  const int mr  = lane & 15;                 // A row / B col / D col index
  const int klo = (lane >> 4) * 2;           // K sub-pair for A and B frags
  const int h   = h0 + mr;

  // B fragments: B[k][n] = M[(wo0+n)][kw+k]  (consecutive k -> v2f load, 8B aligned)
  v2f b0[4], b1[4];
#pragma unroll
  for (int kk = 0; kk < 4; ++kk) {
    const int wi = kw + 4 * kk + klo;
    b0[kk] = *(const v2f*)&M0[(wo0 + mr) * 128 + wi];
    b1[kk] = *(const v2f*)&M1[(wo0 + mr) * 128 + wi];
  }

  // A fragments: A[m][k] = In[h0+m][kw+k]
  const size_t img = (size_t)bc * 128 * 128;
  v2f aYl[4], aHL[4], aLH[4], aHH[4];
#pragma unroll
  for (int kk = 0; kk < 4; ++kk) {
    const int wi = kw + 4 * kk + klo;        // even -> 8B aligned
    aYl[kk] = *(const v2f*)&Yl[img + (size_t)h * 128 + wi];
    aLH[kk] = c2q_pair(Yhr, Yhi, bc, 0, 5, h, wi);
    aHL[kk] = c2q_pair(Yhr, Yhi, bc, 2, 3, h, wi);
    aHH[kk] = c2q_pair(Yhr, Yhi, bc, 1, 4, h, wi);
  }

  v8f c1 = {}, c2 = {};
#pragma unroll
  for (int kk = 0; kk < 4; ++kk) c1 = WMMA_F32(aYl[kk], b0[kk], c1);
#pragma unroll
  for (int kk = 0; kk < 4; ++kk) c1 = WMMA_F32(aHL[kk], b1[kk], c1);
#pragma unroll
  for (int kk = 0; kk < 4; ++kk) c2 = WMMA_F32(aLH[kk], b0[kk], c2);
#pragma unroll
  for (int kk = 0; kk < 4; ++kk) c2 = WMMA_F32(aHH[kk], b1[kk], c2);

  // D layout: VGPR v -> M = v + (lane>=16)*8, N = lane&15
  const size_t ob = (size_t)bc * 128 * 256;
  const int col = wo0 + mr;
  const int mb  = (lane >> 4) * 8;
#pragma unroll
  for (int v = 0; v < 8; ++v) {
    const int row = h0 + mb + v;
    Y1[ob + (size_t)row * 256 + col] = c1[v];
    Y2[ob + (size_t)row * 256 + col] = c2[v];
  }
}

// ---------------------------------------------------------------------------
// Pass 2: column filter.  out[ho,wo] = sum_hi M0[ho,hi]*y1[hi,wo]
//                                           + M1[ho,hi]*y2[hi,wo]
// ---------------------------------------------------------------------------
__global__ __launch_bounds__(256)
void pass2_colfilt(const float* __restrict__ Y1, const float* __restrict__ Y2,
                   const float* __restrict__ M0, const float* __restrict__ M1,
                   float* __restrict__ Out) {
  const int lane = threadIdx.x & 31;
  const int wv   = threadIdx.x >> 5;
  const int tile = blockIdx.x * 8 + wv;      // 512*16*16 tiles total
  const int wob = tile & 15;
  const int hob = (tile >> 4) & 15;
  const int bc  = tile >> 8;                 // 0..511
  const int wo0 = wob * 16, ho0 = hob * 16;
  int kw = ho0 / 2 - 4;
  kw = kw < 0 ? 0 : (kw > 112 ? 112 : kw);

  const int mr  = lane & 15;
  const int klo = (lane >> 4) * 2;

  // A fragments: A[m][k] = M[(ho0+m)][kw+k]
  v2f a0[4], a1[4];
#pragma unroll
  for (int kk = 0; kk < 4; ++kk) {
    const int hi = kw + 4 * kk + klo;
    a0[kk] = *(const v2f*)&M0[(ho0 + mr) * 128 + hi];
    a1[kk] = *(const v2f*)&M1[(ho0 + mr) * 128 + hi];
  }

  // B fragments: B[k][n] = Y[(kw+k)][wo0+n]  (k -> different rows, coalesced over lanes)
  const size_t base = (size_t)bc * 128 * 256;
  v2f bb1[4], bb2[4];
#pragma unroll
  for (int kk = 0; kk < 4; ++kk) {
    const int kb = kw + 4 * kk + klo;
    const int col = wo0 + mr;
    bb1[kk].x = Y1[base + (size_t)kb * 256 + col];
    bb1[kk].y = Y1[base + (size_t)(kb + 1) * 256 + col];
    bb2[kk].x = Y2[base + (size_t)kb * 256 + col];
    bb2[kk].y = Y2[base + (size_t)(kb + 1) * 256 + col];
  }

  v8f c = {};
#pragma unroll
  for (int kk = 0; kk < 4; ++kk) c = WMMA_F32(a0[kk], bb1[kk], c);
#pragma unroll
  for (int kk = 0; kk < 4; ++kk) c = WMMA_F32(a1[kk], bb2[kk], c);

  const size_t ob = (size_t)bc * 256 * 256;
  const int col = wo0 + mr;
  const int mb  = (lane >> 4) * 8;
#pragma unroll
  for (int v = 0; v < 8; ++v) {
    Out[ob + (size_t)(ho0 + mb + v) * 256 + col] = c[v];
  }
}

// ---------------------------------------------------------------------------
// Launch.  Workspace layout (floats):
//   [0,       32768)  M0 (256x128)
//   [32768,   65536)  M1 (256x128)
//   [65536,  +16Mi )  Y1 (512 x 128 x 256)
//   [...,    +16Mi )  Y2 (512 x 128 x 256)
// Total ~128.25 MiB.
// ---------------------------------------------------------------------------
extern "C" void kernel_launch(void* const* d_in, const int* in_sizes, int n_in,
                              void* d_out, int out_size, void* d_ws,
                              size_t ws_size, hipStream_t stream) {
  const float* Yl  = (const float*)d_in[0];
  const float* Yhr = (const float*)d_in[1];
  const float* Yhi = (const float*)d_in[2];
  float* out = (float*)d_out;
  float* ws  = (float*)d_ws;

  float* M0 = ws;
  float* M1 = ws + 32768;
  float* Y1 = ws + 65536;
  float* Y2 = Y1 + (size_t)512 * 128 * 256;

  build_M<<<1, 256, 0, stream>>>(M0, M1);
  // pass1: 512 images * 8 hblocks * 16 woblocks = 65536 tiles, 8 waves/block
  pass1_rowfilt<<<65536 / 8, 256, 0, stream>>>(Yl, Yhr, Yhi, M0, M1, Y1, Y2);
  // pass2: 512 images * 16 * 16 = 131072 tiles, 8 waves/block
  pass2_colfilt<<<131072 / 8, 256, 0, stream>>>(Y1, Y2, M0, M1, out);
}